// LineMSELoss_72524817760731
// MI455X (gfx1250) — compile-verified
//
#include <hip/hip_runtime.h>
#include <hip/hip_bf16.h>

typedef __attribute__((ext_vector_type(2))) float v2f;
typedef __attribute__((ext_vector_type(8))) float v8f;

#define BB 32
#define KK 17
#define HH 256
#define WW 256
#define NE 19   // skeleton edges

__device__ __constant__ int SKEL0[NE] = {15,13,16,14,11, 5, 6, 5, 5, 6, 7, 8, 1, 0, 0, 1, 2, 3, 4};
__device__ __constant__ int SKEL1[NE] = {13,11,14,12,12,11,12, 6, 7, 8, 9,10, 2, 1, 2, 3, 4, 5, 6};

// ---------------------------------------------------------------------------
// Kernel 1: streaming single-pass soft-argmax.
// grid = 2*B*K blocks (one per heatmap slab), 256 threads.
// Thread t reads float4 at (row, 4*(t&63)) for rows (t>>6)+4i -> fully
// coalesced 4KB/step across the block; each input byte read exactly once.
// ---------------------------------------------------------------------------
__global__ void __launch_bounds__(256)
softargmax_kernel(const float* __restrict__ outp, const float* __restrict__ tgtp,
                  float* __restrict__ coords) {
    const int slab  = blockIdx.x;              // [0, 2*B*K)
    const int which = slab / (BB * KK);
    const int bk    = slab - which * (BB * KK);
    const float* src = (which ? tgtp : outp) + (size_t)bk * (HH * WW);

    const int t     = threadIdx.x;
    const int col4  = (t & 63) * 4;
    const int row0  = t >> 6;

    float se = 0.0f, sex = 0.0f, sey = 0.0f;
    #pragma unroll 4
    for (int i = 0; i < 64; ++i) {
        const int y = row0 + i * 4;
        const float4 v = *(const float4*)(src + (size_t)y * WW + col4);
        const float e0 = __expf(v.x);
        const float e1 = __expf(v.y);
        const float e2 = __expf(v.z);
        const float e3 = __expf(v.w);
        const float s  = (e0 + e1) + (e2 + e3);
        se  += s;
        sey = fmaf(s, (float)y, sey);
        sex += fmaf(s, (float)col4, e1 + 2.0f * e2 + 3.0f * e3);
    }

    __shared__ float r0[256], r1[256], r2[256];
    r0[t] = se; r1[t] = sex; r2[t] = sey;
    __syncthreads();
    for (int s = 128; s > 0; s >>= 1) {
        if (t < s) { r0[t] += r0[t + s]; r1[t] += r1[t + s]; r2[t] += r2[t + s]; }
        __syncthreads();
    }
    if (t == 0) {
        const float inv = 1.0f / r0[0];
        coords[slab * 2 + 0] = r1[0] * inv;   // expected x
        coords[slab * 2 + 1] = r2[0] * inv;   // expected y
    }
}

// ---------------------------------------------------------------------------
// Kernel 2: line-gaussian MSE. grid = B*NE blocks, 256 threads = 8 wave32s.
// Each wave evaluates the affine field  f = dy*x - dx*y + c  over a 16x16
// pixel tile with one V_WMMA_F32_16X16X4_F32 per heatmap:
//   A(16x4): A[m] = [1, m, 0, 0]      (lane<16: {1, lane}; lane>=16: 0)
//   B(4x16): B[:,n] = [dy*(tx+n)+c-dx*ty, -dx, 0, 0]
//   D[m][n] = dy*(tx+n) - dx*(ty+m) + c
// Pred/tgt D share the same VGPR layout, so each register slot pairs the same
// pixel; the gaussian + squared diff is pure VALU/trans on top.
// ---------------------------------------------------------------------------
__global__ void __launch_bounds__(256)
line_mse_kernel(const float* __restrict__ coords, float* __restrict__ partials) {
    const int blk = blockIdx.x;               // [0, B*NE)
    const int e   = blk % NE;
    const int b   = blk / NE;
    const int j0  = SKEL0[e];
    const int j1  = SKEL1[e];

    // coords layout: slab = which*B*K + b*K + k, then {x, y}
    const int basep = (0 * BB * KK + b * KK) * 2;
    const int baset = (1 * BB * KK + b * KK) * 2;
    const float x0p = coords[basep + j0 * 2 + 0], y0p = coords[basep + j0 * 2 + 1];
    const float x1p = coords[basep + j1 * 2 + 0], y1p = coords[basep + j1 * 2 + 1];
    const float x0t = coords[baset + j0 * 2 + 0], y0t = coords[baset + j0 * 2 + 1];
    const float x1t = coords[baset + j1 * 2 + 0], y1t = coords[baset + j1 * 2 + 1];

    const float dyp = y1p - y0p, dxp = x1p - x0p, cp = x1p * y0p - y1p * x0p;
    const float dyt = y1t - y0t, dxt = x1t - x0t, ct = x1t * y0t - y1t * x0t;
    const float invp = 1.0f / (sqrtf(dyp * dyp + dxp * dxp) + 1e-6f);
    const float invt = 1.0f / (sqrtf(dyt * dyt + dxt * dxt) + 1e-6f);
    // exp(-dist^2/(2*sigma^2)), sigma=0.5  ->  exp(-2 * f^2 * inv^2)
    const float kfp = -2.0f * invp * invp;
    const float kft = -2.0f * invt * invt;

    const int tid  = threadIdx.x;
    const int lane = tid & 31;
    const int wv   = tid >> 5;
    const bool lo  = (lane < 16);

    v2f a;
    a.x = lo ? 1.0f : 0.0f;          // A[m][0] = 1      (K=2 half zero)
    a.y = lo ? (float)lane : 0.0f;   // A[m][1] = m      (K=3 half zero)
    const v8f zc = {};

    float acc = 0.0f;
    for (int it = 0; it < 32; ++it) {
        const int tile = wv + (it << 3);            // 256 tiles of 16x16
        const float ty = (float)((tile >> 4) << 4);
        const float tx = (float)((tile & 15) << 4);
        const float xl = tx + (float)lane;          // pixel x for column n=lane

        v2f bp, bt;
        bp.x = lo ? fmaf(dyp, xl, cp - dxp * ty) : 0.0f;  // B[0][n]
        bp.y = lo ? -dxp : 0.0f;                          // B[1][n]
        bt.x = lo ? fmaf(dyt, xl, ct - dxt * ty) : 0.0f;
        bt.y = lo ? -dxt : 0.0f;

        v8f fp = __builtin_amdgcn_wmma_f32_16x16x4_f32(
            false, a, false, bp, (short)0, zc, false, false);
        v8f ft = __builtin_amdgcn_wmma_f32_16x16x4_f32(
            false, a, false, bt, (short)0, zc, false, false);

        #pragma unroll
        for (int r = 0; r < 8; ++r) {
            const float fpv = fp[r];
            const float ftv = ft[r];
            const float gp = __expf(kfp * fpv * fpv);
            const float gt = __expf(kft * ftv * ftv);
            const float d  = gp - gt;
            acc = fmaf(d, d, acc);
        }
    }

    __shared__ float red[256];
    red[tid] = acc;
    __syncthreads();
    for (int s = 128; s > 0; s >>= 1) {
        if (tid < s) red[tid] += red[tid + s];
        __syncthreads();
    }
    if (tid == 0) partials[blk] = red[0];
}

// ---------------------------------------------------------------------------
// Kernel 3: deterministic final reduction of B*NE partials -> scalar loss.
// ---------------------------------------------------------------------------
__global__ void __launch_bounds__(256)
finalize_kernel(const float* __restrict__ partials, float* __restrict__ out) {
    __shared__ float red[256];
    const int t = threadIdx.x;
    float s = 0.0f;
    for (int i = t; i < BB * NE; i += 256) s += partials[i];
    red[t] = s;
    __syncthreads();
    for (int st = 128; st > 0; st >>= 1) {
        if (t < st) red[t] += red[t + st];
        __syncthreads();
    }
    if (t == 0)
        out[0] = red[0] * (1.0f / ((float)BB * (float)NE * (float)(HH * WW)));
}

extern "C" void kernel_launch(void* const* d_in, const int* in_sizes, int n_in,
                              void* d_out, int out_size, void* d_ws, size_t ws_size,
                              hipStream_t stream) {
    const float* outp = (const float*)d_in[0];
    const float* tgtp = (const float*)d_in[1];
    float* ws       = (float*)d_ws;
    float* coords   = ws;          // 2*B*K*2 = 2176 floats
    float* partials = ws + 4096;   // B*NE   =  608 floats

    softargmax_kernel<<<2 * BB * KK, 256, 0, stream>>>(outp, tgtp, coords);
    line_mse_kernel<<<BB * NE, 256, 0, stream>>>(coords, partials);
    finalize_kernel<<<1, 256, 0, stream>>>(partials, (float*)d_out);
}